// OHEMBootstrappedCrossEntropy2D_3882650436023
// MI455X (gfx1250) — compile-verified
//
#include <hip/hip_runtime.h>
#include <hip/hip_bf16.h>
#include <cstdint>
#include <math.h>

// ---------------- problem constants ----------------
#define N_      8
#define C_      19
#define H_      768
#define W_      768
#define HW_     (H_*W_)
#define OH_     96
#define OW_     96
#define NDS_    (N_*OH_*OW_)          // 73728 downsampled entries
#define IGNORE_IDX (-100)
#define THRESH_    0.7f
#define MIN_KEPT_DS 1562              // 100000 / 64
#define TOPK_      128

// ---------------- ws layout (bytes) ----------------
#define WS_CTRL_OFF (NDS_*4)
#define WS_LOSS_OFF (WS_CTRL_OFF + 256)

// ---------------- TDM availability ----------------
#if __has_builtin(__builtin_amdgcn_tensor_load_to_lds) && __has_builtin(__builtin_amdgcn_s_wait_tensorcnt)
#define USE_TDM 1
#else
#define USE_TDM 0
#endif

typedef unsigned int u32x4 __attribute__((ext_vector_type(4)));
typedef int          i32x8 __attribute__((ext_vector_type(8)));
typedef int          i32x4 __attribute__((ext_vector_type(4)));

#define NREP 8                        // histogram replicas (contention relief)

// =====================================================================
// Kernel 0: init control block
// =====================================================================
__global__ void k_init(unsigned* ctrl) {
    if (threadIdx.x == 0) ctrl[0] = 0u;   // num_valid
}

// =====================================================================
// Kernel 1: downsampled true-class probability
// =====================================================================
__device__ __forceinline__ float softmax_prob_at(const float* __restrict__ base,
                                                 int y, int x, int ct) {
    const float* p = base + (size_t)y * W_ + x;
    float v[C_];
#pragma unroll
    for (int c = 0; c < C_; ++c) v[c] = p[(size_t)c * HW_];
    float m = v[0];
#pragma unroll
    for (int c = 1; c < C_; ++c) m = fmaxf(m, v[c]);
    float s = 0.f, vt = v[0];
#pragma unroll
    for (int c = 0; c < C_; ++c) {
        s += expf(v[c] - m);
        vt = (c == ct) ? v[c] : vt;
    }
    return expf(vt - m) / s;
}

__global__ __launch_bounds__(256) void k_downsample(const float* __restrict__ preds,
                                                    const int*   __restrict__ tgt,
                                                    unsigned*    __restrict__ ds_bits,
                                                    unsigned*    __restrict__ num_valid) {
    __shared__ unsigned s_cnt;
    if (threadIdx.x == 0) s_cnt = 0u;
    __syncthreads();

    int idx = blockIdx.x * blockDim.x + threadIdx.x;   // NDS_ % 256 == 0
    if (idx < NDS_) {
        int n  = idx / (OH_*OW_);
        int r  = idx - n * (OH_*OW_);
        int oy = r / OW_;
        int ox = r - oy * OW_;

        const float sc = (float)(H_ - 1) / (float)(OH_ - 1);   // align-corners
        float fy = oy * sc, fx = ox * sc;
        int   y0 = (int)floorf(fy), x0 = (int)floorf(fx);
        float ay = fy - (float)y0,  ax = fx - (float)x0;
        int   y1 = min(y0 + 1, H_ - 1), x1 = min(x0 + 1, W_ - 1);

        int iy = (int)floorf(fy + 0.5f);                       // nearest target
        int ix = (int)floorf(fx + 0.5f);
        int t  = tgt[(size_t)n * HW_ + (size_t)iy * W_ + ix];
        bool valid = (t != IGNORE_IDX);
        int  ct = min(max(t, 0), C_ - 1);

        float pr = __int_as_float(0x7f800000);                 // +inf sorts last
        if (valid) {
            const float* base = preds + (size_t)n * C_ * HW_;
            float p00 = softmax_prob_at(base, y0, x0, ct);
            float p01 = softmax_prob_at(base, y0, x1, ct);
            float p10 = softmax_prob_at(base, y1, x0, ct);
            float p11 = softmax_prob_at(base, y1, x1, ct);
            pr = (1.f - ay) * ((1.f - ax) * p00 + ax * p01)
               +        ay  * ((1.f - ax) * p10 + ax * p11);
            atomicAdd(&s_cnt, 1u);                             // LDS, per-block
        }
        ds_bits[idx] = __float_as_uint(pr);
    }
    __syncthreads();
    if (threadIdx.x == 0 && s_cnt) atomicAdd(num_valid, s_cnt);  // 1 global atomic/block
}

// =====================================================================
// Kernel 2: exact k-th smallest of 73728 positive floats, 3-pass radix
// select with NREP replicated LDS histograms. Single 1024-thread WG.
// =====================================================================
__global__ __launch_bounds__(1024) void k_threshold(const unsigned* __restrict__ ds_bits,
                                                    const unsigned* __restrict__ num_valid_p,
                                                    float* __restrict__ thr_out) {
    __shared__ unsigned hist[NREP][2048];
    __shared__ unsigned s_prefix, s_pmask, s_remk;
    unsigned tid = threadIdx.x;
    unsigned rep = (tid >> 5) & (NREP - 1);
    if (tid == 0) {
        unsigned nv = *num_valid_p;
        int k = (int)min(nv, (unsigned)MIN_KEPT_DS) - 1;
        s_remk = (unsigned)max(k, 0);
        s_prefix = 0u; s_pmask = 0u;
    }
    __syncthreads();

    const int shifts[3] = {21, 10, 0};
    const int nbits [3] = {11, 11, 10};
#pragma unroll 1
    for (int pass = 0; pass < 3; ++pass) {
        const int shift = shifts[pass];
        const unsigned nb = 1u << nbits[pass], bm = nb - 1u;
        {
            unsigned* hf = &hist[0][0];
            for (unsigned j = tid; j < NREP * 2048u; j += 1024u) hf[j] = 0u;
        }
        __syncthreads();
        const unsigned pre = s_prefix, pm = s_pmask;
        const uint4* vb4 = (const uint4*)ds_bits;
        for (unsigned i = tid; i < (unsigned)(NDS_ / 4); i += 1024u) {   // b128 loads
            uint4 q = vb4[i];
            if ((q.x & pm) == pre) atomicAdd(&hist[rep][(q.x >> shift) & bm], 1u);
            if ((q.y & pm) == pre) atomicAdd(&hist[rep][(q.y >> shift) & bm], 1u);
            if ((q.z & pm) == pre) atomicAdd(&hist[rep][(q.z >> shift) & bm], 1u);
            if ((q.w & pm) == pre) atomicAdd(&hist[rep][(q.w >> shift) & bm], 1u);
        }
        __syncthreads();
        for (unsigned j = tid; j < nb; j += 1024u) {            // fold replicas
            unsigned s = 0;
#pragma unroll
            for (int r = 0; r < NREP; ++r) s += hist[r][j];
            hist[0][j] = s;
        }
        __syncthreads();
        if (tid == 0) {
            unsigned cum = 0, rk = s_remk;
            for (unsigned b = 0; b < nb; ++b) {
                unsigned c = hist[0][b];
                if (cum + c > rk) {
                    s_prefix = pre | (b << shift);
                    s_pmask  = pm  | (bm << shift);
                    s_remk   = rk - cum;
                    break;
                }
                cum += c;
            }
        }
        __syncthreads();
    }
    if (tid == 0) {
        float kth = __uint_as_float(s_prefix);
        float thr = (kth > THRESH_) ? kth : THRESH_;
        unsigned nv = *num_valid_p;
        if ((unsigned)MIN_KEPT_DS >= nv) thr = 1.0f;
        *thr_out = thr;
    }
}

// =====================================================================
// Kernel 3: main 359MB pass. Per-wave TDM double-buffer pipeline:
// D# tile = 19 channels x 32 pixels, stride H*W, global -> LDS.
// =====================================================================
#define TILE_PX 32
#define ITER_   16
#define WPB_    8
#define TILES_TOTAL  (N_*HW_/TILE_PX)             // 147456
#define MAIN_BLOCKS  (TILES_TOTAL/(ITER_*WPB_))   // 1152

#if USE_TDM
__device__ __forceinline__ void tdm_issue_tile(const float* __restrict__ preds,
                                               int tile, unsigned lds_byte_off) {
    asm volatile("s_wait_dscnt 0" ::: "memory");  // drain LDS reads of reused buffer
    int n = tile / (HW_ / TILE_PX);
    unsigned long long elem = (unsigned long long)tile * TILE_PX
                            + (unsigned long long)n * (unsigned long long)(C_ - 1) * HW_;
    unsigned long long ga = (unsigned long long)(uintptr_t)preds + elem * 4ull;

    u32x4 g0;
    g0.x = 1u;                                        // count=1 (valid descriptor)
    g0.y = lds_byte_off;                              // lds_addr
    g0.z = (unsigned)(ga & 0xFFFFFFFFull);            // global_addr[31:0]
    g0.w = (unsigned)((ga >> 32) & 0x1FFFFFFull)      // global_addr[56:32]
         | (2u << 30);                                // type=2 ("image")

    i32x8 g1;
    g1[0] = (int)(2u << 16);                          // data_size=2 -> 4 bytes
    g1[1] = (int)((unsigned)TILE_PX << 16);           // tensor_dim0=32
    g1[2] = (int)((unsigned)C_ << 16);                // tensor_dim1=19
    g1[3] = (int)((unsigned)TILE_PX << 16);           // tile_dim0=32
    g1[4] = C_;                                       // tile_dim1=19, tile_dim2=0
    g1[5] = HW_;                                      // tensor_dim0_stride = 589824
    g1[6] = 0;
    g1[7] = 0;
    i32x4 z4 = {0, 0, 0, 0};
    i32x8 z8 = {0, 0, 0, 0, 0, 0, 0, 0};
    __builtin_amdgcn_tensor_load_to_lds(g0, g1, z4, z4, z8, 0);
}
#endif

__global__ __launch_bounds__(256) void k_mainloss(const float* __restrict__ preds,
                                                  const int*   __restrict__ tgt,
                                                  const float* __restrict__ wgt,
                                                  const float* __restrict__ thr_p,
                                                  float*       __restrict__ loss) {
    __shared__ float swgt[C_];
#if USE_TDM
    __shared__ float smem[WPB_][2][C_ * TILE_PX];     // 38,912 B of 320 KB
#endif
    if (threadIdx.x < C_) swgt[threadIdx.x] = wgt[threadIdx.x];
    __syncthreads();

    const int lane = threadIdx.x & 31;
    const int wid  = threadIdx.x >> 5;
    const float thr = thr_p[0];

    int tile0 = __builtin_amdgcn_readfirstlane((blockIdx.x * WPB_ + wid) * ITER_);

#if USE_TDM
    unsigned lds_base = __builtin_amdgcn_readfirstlane(
        (unsigned)(uintptr_t)&smem[wid][0][0]);
    const unsigned buf_stride = (unsigned)(C_ * TILE_PX * 4);

    tdm_issue_tile(preds, tile0, lds_base);           // prime the pipe
#pragma unroll 1
    for (int i = 0; i < ITER_; ++i) {
        if (i + 1 < ITER_) {
            tdm_issue_tile(preds, tile0 + i + 1, lds_base + ((unsigned)((i + 1) & 1)) * buf_stride);
            __builtin_amdgcn_s_wait_tensorcnt((short)1);   // oldest tile landed
        } else {
            __builtin_amdgcn_s_wait_tensorcnt((short)0);
        }
        const float* buf = &smem[wid][i & 1][0];
        float v[C_];
#pragma unroll
        for (int c = 0; c < C_; ++c) v[c] = buf[c * TILE_PX + lane];
#else
#pragma unroll 1
    for (int i = 0; i < ITER_; ++i) {
        float v[C_];
#endif
        size_t pix = (size_t)(tile0 + i) * TILE_PX + lane;   // [0, N*HW)
#if !USE_TDM
        {
            int n = (int)(pix / HW_);
            size_t e = pix + (size_t)n * (size_t)(C_ - 1) * HW_;
#pragma unroll
            for (int c = 0; c < C_; ++c) v[c] = preds[e + (size_t)c * HW_];
        }
#endif
        int t = tgt[pix];
        bool valid = (t != IGNORE_IDX);
        int ct = min(max(t, 0), C_ - 1);

        float m = v[0];
#pragma unroll
        for (int c = 1; c < C_; ++c) m = fmaxf(m, v[c]);
        float s = 0.f, vt = v[0];
#pragma unroll
        for (int c = 0; c < C_; ++c) {
            s += expf(v[c] - m);
            vt = (c == ct) ? v[c] : vt;
        }
        float lp = vt - m - logf(s);       // log-softmax at true class
        float p  = expf(lp);               // softmax prob at true class
        float l  = -lp * swgt[ct];
        loss[pix] = (valid && p <= thr) ? l : 0.f;
    }
}

// =====================================================================
// Kernel 4: per-sample exact top-128 sum: descending radix select with
// replicated histograms, then tie-exact deterministic sum.
// =====================================================================
__global__ __launch_bounds__(1024) void k_topk(const float* __restrict__ loss,
                                               float* __restrict__ sums) {
    const int n = blockIdx.x;
    const float*    vf = loss + (size_t)n * HW_;
    const unsigned* vb = (const unsigned*)vf;

    __shared__ unsigned hist[NREP][2048];
    __shared__ unsigned s_prefix, s_pmask, s_remk;
    __shared__ float    red [1024];
    __shared__ unsigned redc[1024];

    unsigned tid = threadIdx.x;
    unsigned rep = (tid >> 5) & (NREP - 1);
    if (tid == 0) { s_remk = TOPK_ - 1; s_prefix = 0u; s_pmask = 0u; }
    __syncthreads();

    const int shifts[3] = {21, 10, 0};
    const int nbits [3] = {11, 11, 10};
#pragma unroll 1
    for (int pass = 0; pass < 3; ++pass) {
        const int shift = shifts[pass];
        const unsigned nb = 1u << nbits[pass], bm = nb - 1u;
        {
            unsigned* hf = &hist[0][0];
            for (unsigned j = tid; j < NREP * 2048u; j += 1024u) hf[j] = 0u;
        }
        __syncthreads();
        const unsigned pre = s_prefix, pm = s_pmask;
        const uint4* vb4 = (const uint4*)vb;
        for (unsigned i = tid; i < (unsigned)(HW_ / 4); i += 1024u) {    // b128 loads
            uint4 q = vb4[i];
            if ((q.x & pm) == pre) atomicAdd(&hist[rep][(q.x >> shift) & bm], 1u);
            if ((q.y & pm) == pre) atomicAdd(&hist[rep][(q.y >> shift) & bm], 1u);
            if ((q.z & pm) == pre) atomicAdd(&hist[rep][(q.z >> shift) & bm], 1u);
            if ((q.w & pm) == pre) atomicAdd(&hist[rep][(q.w >> shift) & bm], 1u);
        }
        __syncthreads();
        for (unsigned j = tid; j < nb; j += 1024u) {            // fold replicas
            unsigned s = 0;
#pragma unroll
            for (int r = 0; r < NREP; ++r) s += hist[r][j];
            hist[0][j] = s;
        }
        __syncthreads();
        if (tid == 0) {
            unsigned cum = 0, rk = s_remk;
            for (int b = (int)nb - 1; b >= 0; --b) {            // descending
                unsigned c = hist[0][b];
                if (cum + c > rk) {
                    s_prefix = pre | ((unsigned)b << shift);
                    s_pmask  = pm  | (bm << shift);
                    s_remk   = rk - cum;
                    break;
                }
                cum += c;
            }
        }
        __syncthreads();
    }

    const float cutoff = __uint_as_float(s_prefix);   // 128th largest (exact bits)
    float    ps = 0.f;
    unsigned pc = 0;
    const float4* vf4 = (const float4*)vf;
    for (unsigned i = tid; i < (unsigned)(HW_ / 4); i += 1024u) {
        float4 q = vf4[i];
        if (q.x > cutoff) { ps += q.x; pc++; }
        if (q.y > cutoff) { ps += q.y; pc++; }
        if (q.z > cutoff) { ps += q.z; pc++; }
        if (q.w > cutoff) { ps += q.w; pc++; }
    }
    red[tid] = ps; redc[tid] = pc;
    __syncthreads();
    for (unsigned s = 512; s > 0; s >>= 1) {          // fixed-order => deterministic
        if (tid < s) { red[tid] += red[tid + s]; redc[tid] += redc[tid + s]; }
        __syncthreads();
    }
    if (tid == 0)
        sums[n] = red[0] + cutoff * (float)(TOPK_ - (int)redc[0]);
}

// =====================================================================
// Kernel 5: deterministic scalar combine
// =====================================================================
__global__ void k_final(const float* __restrict__ sums, float* __restrict__ out) {
    if (threadIdx.x == 0 && blockIdx.x == 0) {
        float acc = 0.f;
#pragma unroll
        for (int i = 0; i < N_; ++i) acc += sums[i] * (1.0f / (float)TOPK_);
        out[0] = acc / (float)N_;
    }
}

// =====================================================================
extern "C" void kernel_launch(void* const* d_in, const int* in_sizes, int n_in,
                              void* d_out, int out_size, void* d_ws, size_t ws_size,
                              hipStream_t stream) {
    (void)in_sizes; (void)n_in; (void)out_size; (void)ws_size;
    const float* preds = (const float*)d_in[0];          // [8,19,768,768] f32
    const float* wgt   = (const float*)d_in[1];          // [19] f32
    const int*   tgt   = (const int*)  d_in[2];          // [8,768,768] int
    float* out = (float*)d_out;

    char* ws = (char*)d_ws;
    unsigned* ds_bits = (unsigned*)(ws);
    unsigned* ctrl    = (unsigned*)(ws + WS_CTRL_OFF);
    float*    ctrlf   = (float*)ctrl;                    // [1]=thr, [2..9]=sample sums
    float*    loss    = (float*)(ws + WS_LOSS_OFF);

    k_init<<<1, 1, 0, stream>>>(ctrl);
    k_downsample<<<(NDS_ + 255) / 256, 256, 0, stream>>>(preds, tgt, ds_bits, ctrl);
    k_threshold<<<1, 1024, 0, stream>>>(ds_bits, ctrl, ctrlf + 1);
    k_mainloss<<<MAIN_BLOCKS, 256, 0, stream>>>(preds, tgt, wgt, ctrlf + 1, loss);
    k_topk<<<N_, 1024, 0, stream>>>(loss, ctrlf + 2);
    k_final<<<1, 1, 0, stream>>>(ctrlf + 2, out);
}